// GraphConv_17008070492258
// MI455X (gfx1250) — compile-verified
//
#include <hip/hip_runtime.h>

typedef float v2f __attribute__((ext_vector_type(2)));
typedef float v8f __attribute__((ext_vector_type(8)));

#define D_FEAT 128

// ---------------- K1: degree counts (float, via atomics) ----------------
__global__ __launch_bounds__(256) void degree_kernel(const int* __restrict__ src,
                                                     const int* __restrict__ dst,
                                                     float* __restrict__ deg_out,
                                                     float* __restrict__ deg_in,
                                                     int n_edges) {
    int e = blockIdx.x * blockDim.x + threadIdx.x;
    if (e >= n_edges) return;
    atomicAdd(&deg_out[src[e]], 1.0f);
    atomicAdd(&deg_in[dst[e]], 1.0f);
}

// ---------------- K2: deg -> clip(.,1)^-1/2 scales ----------------
__global__ __launch_bounds__(256) void scale_kernel(const float* __restrict__ deg_out,
                                                    const float* __restrict__ deg_in,
                                                    float* __restrict__ s_out,
                                                    float* __restrict__ s_in,
                                                    int n_nodes) {
    int i = blockIdx.x * blockDim.x + threadIdx.x;
    if (i >= n_nodes) return;
    float o = deg_out[i]; o = (o < 1.0f) ? 1.0f : o;
    float d = deg_in[i];  d = (d < 1.0f) ? 1.0f : d;
    s_out[i] = 1.0f / sqrtf(o);
    s_in[i]  = 1.0f / sqrtf(d);
}

// ---------------- K3: edge scatter-add (one wave per edge) ----------------
// agg[dst[e], :] += feat[src[e], :] * s_out[src[e]]
__global__ __launch_bounds__(256) void scatter_kernel(const float* __restrict__ feat,
                                                      const int* __restrict__ src,
                                                      const int* __restrict__ dst,
                                                      const float* __restrict__ s_out,
                                                      float* __restrict__ agg,
                                                      int n_edges) {
    int wavesPerBlock = blockDim.x >> 5;
    int e = blockIdx.x * wavesPerBlock + (threadIdx.x >> 5);
    int lane = threadIdx.x & 31;
    if (e >= n_edges) return;

    int s = src[e];
    int d = dst[e];
    float sc = s_out[s];

    // 32 lanes x float4 = 128 floats (one full feature row), coalesced b128 load
    const float4* fp = (const float4*)(feat + (size_t)s * D_FEAT);
    float4 v = fp[lane];

    float* op = agg + (size_t)d * D_FEAT + (size_t)lane * 4;
    atomicAdd(op + 0, v.x * sc);
    atomicAdd(op + 1, v.y * sc);
    atomicAdd(op + 2, v.z * sc);
    atomicAdd(op + 3, v.w * sc);
}

// ---------------- K4: in-place fp32 WMMA GEMM + dst-side scaling ----------------
// out[r, :] = (out[r, :] @ W) * s_in[r], done per 16-row block.
// Each block: stage 16x128 A-tile in LDS, 8 waves each compute one 16x16 tile
// with 32 x V_WMMA_F32_16X16X4_F32 along K, then overwrite the same rows.
__global__ __launch_bounds__(256) void gemm_scale_kernel(const float* __restrict__ W,
                                                         const float* __restrict__ s_in,
                                                         float* __restrict__ out,
                                                         int n_nodes) {
    __shared__ float Atile[16 * D_FEAT]; // 8 KB

    const int r0  = blockIdx.x * 16;
    const int tid = threadIdx.x;

    // Cooperative tile load: 16 rows x 128 floats = 512 float4, 256 threads
    {
        const float4* src4 = (const float4*)(out + (size_t)r0 * D_FEAT);
        float4* dst4 = (float4*)Atile;
        #pragma unroll
        for (int idx = tid; idx < 512; idx += 256) {
            int row = r0 + (idx >> 5);            // 32 float4 per row
            float4 v;
            if (row < n_nodes) v = src4[idx];
            else { v.x = v.y = v.z = v.w = 0.0f; }
            dst4[idx] = v;
        }
    }
    __syncthreads();

    const int wave  = tid >> 5;          // 0..7  -> N tile = wave*16
    const int lane  = tid & 31;
    const int m     = lane & 15;         // row within tile (A) / col (B,D)
    const int khalf = (lane >> 4) << 1;  // 0 for lanes 0-15, 2 for lanes 16-31
    const int n     = (wave << 4) + m;   // output column

    v8f acc = {};
    #pragma unroll
    for (int k0 = 0; k0 < D_FEAT; k0 += 4) {
        v2f a, b;
        const float* ap = &Atile[m * D_FEAT + k0 + khalf];
        a.x = ap[0];          // A[m][k0+khalf]
        a.y = ap[1];          // A[m][k0+khalf+1]
        const float* bp = &W[(k0 + khalf) * D_FEAT + n];
        b.x = bp[0];          // W[k0+khalf    ][n]
        b.y = bp[D_FEAT];     // W[k0+khalf + 1][n]
        acc = __builtin_amdgcn_wmma_f32_16x16x4_f32(
            /*neg_a=*/false, a, /*neg_b=*/false, b,
            /*c_mod=*/(short)0, acc, /*reuse_a=*/false, /*reuse_b=*/false);
    }

    // D component i -> row r0 + i + 8*(lane/16), column n
    const int rbase = r0 + ((lane >> 4) << 3);
    #pragma unroll
    for (int i = 0; i < 8; ++i) {
        int row = rbase + i;
        if (row < n_nodes) {
            out[(size_t)row * D_FEAT + n] = acc[i] * s_in[row];
        }
    }
}

extern "C" void kernel_launch(void* const* d_in, const int* in_sizes, int n_in,
                              void* d_out, int out_size, void* d_ws, size_t ws_size,
                              hipStream_t stream) {
    const float* feat   = (const float*)d_in[0];   // [N, 128] fp32
    const float* weight = (const float*)d_in[1];   // [128, 128] fp32
    const int*   src    = (const int*)d_in[2];     // [E] int
    const int*   dst    = (const int*)d_in[3];     // [E] int

    const int n_nodes = in_sizes[0] / D_FEAT;
    const int n_edges = in_sizes[2];

    float* out = (float*)d_out;

    // Workspace layout: deg_out | deg_in | s_out | s_in  (4 * n_nodes floats)
    float* ws      = (float*)d_ws;
    float* deg_out = ws;
    float* deg_in  = ws + n_nodes;
    float* s_out   = ws + 2 * (size_t)n_nodes;
    float* s_in    = ws + 3 * (size_t)n_nodes;

    // K0: zero the accumulator (d_out) and the degree counters
    hipMemsetAsync(out, 0, (size_t)out_size * sizeof(float), stream);
    hipMemsetAsync(deg_out, 0, 2 * (size_t)n_nodes * sizeof(float), stream);

    // K1: degrees
    {
        int blocks = (n_edges + 255) / 256;
        degree_kernel<<<blocks, 256, 0, stream>>>(src, dst, deg_out, deg_in, n_edges);
    }
    // K2: scales
    {
        int blocks = (n_nodes + 255) / 256;
        scale_kernel<<<blocks, 256, 0, stream>>>(deg_out, deg_in, s_out, s_in, n_nodes);
    }
    // K3: scatter-add into d_out (one wave per edge)
    {
        int wavesPerBlock = 8; // 256 threads
        int blocks = (n_edges + wavesPerBlock - 1) / wavesPerBlock;
        scatter_kernel<<<blocks, 256, 0, stream>>>(feat, src, dst, s_out, out, n_edges);
    }
    // K4: in-place WMMA GEMM + destination scaling
    {
        int blocks = (n_nodes + 15) / 16;
        gemm_scale_kernel<<<blocks, 256, 0, stream>>>(weight, s_in, out, n_nodes);
    }
}